// embE3Conv_30408368455707
// MI455X (gfx1250) — compile-verified
//
#include <hip/hip_runtime.h>
#include <math.h>
#include <stdint.h>

// ---------------------------------------------------------------------------
// embE3Conv for MI455X (gfx1250, wave32, WMMA f32 16x16x4)
//
// out[e,:9] = sum_q  w[e, P[q]] * sh[e, J[q]] * T2[q, :9]
//   w = silu(emb @ W1/sqrt(8)) * SILU_C @ (W2/sqrt(32))
//   T2[q,k] = alpha_p * sum_i C_p[i, j, k] * f_in[0, o1+i]   (q <-> (p, j))
//
// One wave = one 16-atom tile. Lanes m and m+16 both own atom m so that the
// f32 WMMA A-operand layout (lane m: K=k0,k0+1 / lane m+16: K=k0+2,k0+3)
// needs no cross-lane shuffles. 23 V_WMMA_F32_16X16X4_F32 per tile.
// ---------------------------------------------------------------------------

typedef float v2f __attribute__((ext_vector_type(2)));
typedef float v8f __attribute__((ext_vector_type(8)));
typedef int   v4i __attribute__((ext_vector_type(4)));

#define AS1 __attribute__((address_space(1)))
#define AS3 __attribute__((address_space(3)))

#define NBASIS   8
#define EMBD     32
#define NW       13
#define QROWS    44            // 43 active (p,j) rows + 1 zero pad
#define NCHUNK   (QROWS / 4)   // 11 K-chunks for the final WMMA
#define DIMOUT   9

// d_ws layout (floats)
#define W1B_OFF  0             // 2 ntile * 2 chunk * 32 lane * 2 = 256
#define W2B_OFF  256           // 8 chunk * 32 lane * 2          = 512
#define T2B_OFF  768           // 11 chunk * 32 lane * 2         = 704
#define WS_FLOATS 1472

#if __has_builtin(__builtin_amdgcn_wmma_f32_16x16x4_f32)
#define HAVE_WMMA_F32X4 1
#endif
#if __has_builtin(__builtin_amdgcn_global_load_async_to_lds_b128)
#define HAVE_ASYNC_LDS 1
#endif

__device__ __forceinline__ v8f wmma4(v2f a, v2f b, v8f c) {
#ifdef HAVE_WMMA_F32X4
  return __builtin_amdgcn_wmma_f32_16x16x4_f32(false, a, false, b, (short)0, c,
                                               false, false);
#else
  c[0] += a.x * b.x + a.y * b.y;   // keeps TU compiling; histogram flags it
  return c;
#endif
}

// fast reciprocal: single v_rcp_f32 instead of an IEEE div expansion
__device__ __forceinline__ float frcp(float x) {
  return __builtin_amdgcn_rcpf(x);
}

// ----- path metadata (PATHS order from the reference) ----------------------
__constant__ int   c_o1[NW]    = {0,0,0,1,1,1,1,4,4,4,4,9,9};
__constant__ int   c_o3[NW]    = {0,1,4,1,0,4,1,4,1,0,4,4,1};
__constant__ float c_alpha[NW] = {0.57735027f,0.77459667f,1.f,0.77459667f,
                                  0.57735027f,1.f,0.77459667f,1.f,0.77459667f,
                                  0.57735027f,1.f,1.f,0.77459667f};
__constant__ int   c_qbase[NW] = {0,1,4,9,10,13,16,21,22,25,30,35,38};

// q -> (path, global sh index) for the compacted 43-row G matrix (row 43 pad)
constexpr int QP[QROWS] = {0, 1,1,1, 2,2,2,2,2, 3, 4,4,4, 5,5,5,
                           6,6,6,6,6, 7, 8,8,8, 9,9,9,9,9,
                           10,10,10,10,10, 11,11,11, 12,12,12,12,12, 0};
constexpr int QJ[QROWS] = {0, 1,2,3, 4,5,6,7,8, 0, 1,2,3, 1,2,3,
                           4,5,6,7,8, 0, 1,2,3, 4,5,6,7,8,
                           4,5,6,7,8, 1,2,3, 4,5,6,7,8, 0};

// ----- sparse Wigner (w3j) records: value at C_p[i][j][k] -------------------
// Host-precomputed, seed-0 / largest-entry-positive convention.
#define R(p,i,j,k) (unsigned short)(((p)<<12)|((i)<<8)|((j)<<4)|(k))
#define NREC 96
__constant__ unsigned short c_rec_idx[NREC] = {
  R(0,0,0,0),
  R(1,0,0,0),R(1,0,1,1),R(1,0,2,2),
  R(2,0,0,0),R(2,0,1,1),R(2,0,2,2),R(2,0,3,3),R(2,0,4,4),
  R(3,0,0,0),R(3,1,0,1),R(3,2,0,2),
  R(4,0,0,0),R(4,1,1,0),R(4,2,2,0),
  R(5,0,2,0),R(5,2,0,0),R(5,0,1,1),R(5,1,0,1),R(5,1,1,2),R(5,0,0,2),
  R(5,2,2,2),R(5,1,2,3),R(5,2,1,3),R(5,2,2,4),R(5,0,0,4),
  R(6,0,0,2),R(6,2,0,0),R(6,0,1,1),R(6,1,1,0),R(6,1,2,1),R(6,0,2,0),
  R(6,2,2,2),R(6,1,3,2),R(6,2,3,1),R(6,2,4,2),R(6,0,4,0),
  R(7,0,0,0),R(7,1,0,1),R(7,2,0,2),R(7,3,0,3),R(7,4,0,4),
  R(8,0,0,2),R(8,0,2,0),R(8,1,0,1),R(8,1,1,0),R(8,2,1,1),R(8,2,0,0),
  R(8,2,2,2),R(8,3,1,2),R(8,3,2,1),R(8,4,2,2),R(8,4,0,0),
  R(9,0,0,0),R(9,1,1,0),R(9,2,2,0),R(9,3,3,0),R(9,4,4,0),
  R(10,2,2,2),R(10,0,0,2),R(10,2,0,0),R(10,0,2,0),R(10,3,3,2),R(10,2,3,3),
  R(10,3,2,3),R(10,4,4,2),R(10,2,4,4),R(10,4,2,4),R(10,0,1,3),R(10,1,3,0),
  R(10,3,0,1),R(10,1,1,2),
  R(11,0,0,0),R(11,1,1,0),R(11,2,2,1),R(11,3,1,2),R(11,4,2,2),R(11,1,0,1),
  R(11,5,2,3),R(11,6,0,4),R(11,0,2,4),R(11,2,1,3),R(11,4,0,3),R(11,6,1,1),
  R(12,0,0,0),R(12,1,0,1),R(12,2,1,2),R(12,3,2,1),R(12,4,2,2),R(12,1,1,0),
  R(12,5,3,2),R(12,6,4,0),R(12,0,4,2),R(12,2,3,1),R(12,4,3,0),R(12,6,1,1)
};
__constant__ float c_rec_val[NREC] = {
  1.0f,
  0.57735027f,0.57735027f,0.57735027f,
  0.44721360f,0.44721360f,0.44721360f,0.44721360f,0.44721360f,
  0.57735027f,0.57735027f,0.57735027f,
  0.57735027f,0.57735027f,0.57735027f,
  0.31622777f,0.31622777f,0.31622777f,0.31622777f,0.36514837f,-0.18257419f,
  -0.18257419f,0.31622777f,0.31622777f,0.31622777f,-0.31622777f,
  0.31622777f,0.31622777f,0.31622777f,0.31622777f,0.36514837f,-0.18257419f,
  -0.18257419f,0.31622777f,0.31622777f,0.31622777f,-0.31622777f,
  0.44721360f,0.44721360f,0.44721360f,0.44721360f,0.44721360f,
  0.31622777f,0.31622777f,0.31622777f,0.31622777f,0.36514837f,-0.18257419f,
  -0.18257419f,0.31622777f,0.31622777f,0.31622777f,-0.31622777f,
  0.44721360f,0.44721360f,0.44721360f,0.44721360f,0.44721360f,
  0.29880715f,-0.14940358f,-0.14940358f,-0.14940358f,-0.14940358f,
  -0.14940358f,-0.14940358f,-0.29880715f,-0.29880715f,-0.29880715f,
  0.23904572f,0.23904572f,0.23904572f,0.29880715f,
  0.35355339f,0.25f,0.30618622f,0.35355339f,0.25f,0.30618622f,
  0.35355339f,0.35355339f,-0.25f,0.25f,-0.30618622f,-0.25f,
  0.35355339f,0.25f,0.30618622f,0.35355339f,0.25f,0.30618622f,
  0.35355339f,0.35355339f,-0.25f,0.25f,-0.30618622f,-0.25f
};

// ---------------------------------------------------------------------------
// Prep: build T2 from (w3j ⊗ src), then swizzle W1/W2/T2 into WMMA B-layout.
// ---------------------------------------------------------------------------
__global__ __launch_bounds__(256) void tp_prep(const float* __restrict__ f_in,
                                               const float* __restrict__ w1,
                                               const float* __restrict__ w2,
                                               float* __restrict__ ws) {
  __shared__ float sT2[QROWS * 16];  // 44 x 16 (cols 9..15 + row 43 stay 0)
  const int t = threadIdx.x;
  for (int i = t; i < QROWS * 16; i += 256) sT2[i] = 0.f;
  __syncthreads();
  for (int rI = t; rI < NREC; rI += 256) {
    const unsigned v = c_rec_idx[rI];
    const int p = v >> 12, i = (v >> 8) & 15, jl = (v >> 4) & 15, kl = v & 15;
    const int q = c_qbase[p] + jl;
    const int k = c_o3[p] + kl;
    atomicAdd(&sT2[q * 16 + k], c_alpha[p] * c_rec_val[rI] * f_in[c_o1[p] + i]);
  }
  __syncthreads();

  const float RS8 = 0.35355339f;      // 1/sqrt(NB)
  const float RS32 = 0.17677670f;     // 1/sqrt(EMB)
  const float SILU_C = 1.6791788f;    // second-moment silu normalizer -> W2
  const bool hi32 = (t & 31) >= 16;

  // W1 B-layout: [ntile][chunk][lane] -> float2 (rows k0,k0+1 | k0+2,k0+3)
  for (int i = t; i < 128; i += 256) {
    const int nt = i >> 6, ch = (i >> 5) & 1, lane = i & 31;
    const int n = nt * 16 + (lane & 15);
    const int r0 = ch * 4 + (((lane & 31) >= 16) ? 2 : 0);
    ws[W1B_OFF + 2 * i]     = w1[r0 * EMBD + n] * RS8;
    ws[W1B_OFF + 2 * i + 1] = w1[(r0 + 1) * EMBD + n] * RS8;
  }
  // W2 B-layout (SILU_C folded; cols >= 13 zero)
  {
    const int i = t;  // 256 entries, 256 threads
    const int ch = i >> 5, lane = i & 31;
    const int n = lane & 15;
    const int r0 = ch * 4 + (hi32 ? 2 : 0);
    ws[W2B_OFF + 2 * i]     = (n < NW) ? w2[r0 * NW + n] * RS32 * SILU_C : 0.f;
    ws[W2B_OFF + 2 * i + 1] = (n < NW) ? w2[(r0 + 1) * NW + n] * RS32 * SILU_C : 0.f;
  }
  // T2 B-layout
  for (int i = t; i < NCHUNK * 32; i += 256) {
    const int ch = i >> 5, lane = i & 31;
    const int n = lane & 15;
    const int q0 = ch * 4 + (((lane & 31) >= 16) ? 2 : 0);
    ws[T2B_OFF + 2 * i]     = sT2[q0 * 16 + n];
    ws[T2B_OFF + 2 * i + 1] = sT2[(q0 + 1) * 16 + n];
  }
}

// ---------------------------------------------------------------------------
// Main kernel: 8 waves/block, each wave = one 16-atom tile.
// ---------------------------------------------------------------------------
__global__ __launch_bounds__(256) void tp_conv(const float* __restrict__ pos,
                                               const float* __restrict__ ws,
                                               float* __restrict__ out,
                                               int N) {
  __shared__ __align__(16) float sT2B[NCHUNK * 64];  // 704 f: final-matmul B
  __shared__ __align__(16) float sScr[8][512];       // per-wave h / w scratch

  const int tid  = threadIdx.x;
  const int wave = tid >> 5;
  const int lane = tid & 31;
  const bool hi  = lane >= 16;
  const int m    = lane & 15;            // atom-in-tile (lanes m and m+16)
  const int tile = blockIdx.x * 8 + wave;

  // Stage T2 (B-layout) into LDS. Prefer the CDNA5 async-DMA path
  // (GLOBAL_LOAD_ASYNC_TO_LDS_B128, tracked by ASYNCcnt) over a VGPR bounce.
#ifdef HAVE_ASYNC_LDS
  if (tid < NCHUNK * 16) {  // 176 x 16B = 2816B
    AS1 v4i* gp = (AS1 v4i*)(uintptr_t)(ws + T2B_OFF + 4 * tid);
    // generic->LDS: flat LDS addresses carry the LDS offset in addr[31:0]
    AS3 v4i* lp = (AS3 v4i*)(uint32_t)(uintptr_t)(&sT2B[4 * tid]);
    __builtin_amdgcn_global_load_async_to_lds_b128(gp, lp, 0, 0);
  }
#if __has_builtin(__builtin_amdgcn_s_wait_asynccnt)
  __builtin_amdgcn_s_wait_asynccnt(0);
#else
  asm volatile("s_wait_asynccnt 0" ::: "memory");
#endif
#else
  for (int i = tid; i < NCHUNK * 64; i += 256) sT2B[i] = ws[T2B_OFF + i];
#endif

  // B operands held in registers (uniform across tiles)
  v2f bw1[4];
#pragma unroll
  for (int i = 0; i < 4; ++i)
    bw1[i] = ((const v2f*)(ws + W1B_OFF))[i * 32 + lane];
  v2f bw2[8];
#pragma unroll
  for (int i = 0; i < 8; ++i)
    bw2[i] = ((const v2f*)(ws + W2B_OFF))[i * 32 + lane];

  // ---- per-atom scalar stage (duplicated on the lane pair) ----
  const int e = tile * 16 + m;
  float px = 0.f, py = 0.f, pz = 0.f;
  if (e < N) { px = pos[3 * e]; py = pos[3 * e + 1]; pz = pos[3 * e + 2]; }
  const float r = sqrtf(px * px + py * py + pz * pz);
  const float inv = (r > 0.f) ? frcp(r) : 1.f;
  const float x = px * inv, y = py * inv, z = pz * inv;

  float shv[9];
  shv[0] = 1.f;
  shv[1] = 1.7320508f * x; shv[2] = 1.7320508f * y; shv[3] = 1.7320508f * z;
  shv[4] = 3.8729833f * x * z; shv[5] = 3.8729833f * x * y;
  shv[6] = 2.2360680f * (y * y - 0.5f * (x * x + z * z));
  shv[7] = 3.8729833f * y * z;
  shv[8] = 1.9364917f * (z * z - x * x);

  float emb[8];
#pragma unroll
  for (int i = 0; i < 8; ++i) {
    const float d = (r - (2.f / 3.f) * (float)(i + 1)) * 1.5f;  // /step
    emb[i] = __expf(-d * d) * 2.5253814f;                       // sqrt(8)/1.12
  }
  __syncthreads();  // sT2B visible to all waves

  // ---- MLP layer 1: emb(16x8) @ W1(8x32) -> two 16x16 C tiles ----
  v8f c0 = {}, c1 = {};
  {
    v2f a0; a0.x = hi ? emb[2] : emb[0]; a0.y = hi ? emb[3] : emb[1];
    v2f a1; a1.x = hi ? emb[6] : emb[4]; a1.y = hi ? emb[7] : emb[5];
    c0 = wmma4(a0, bw1[0], c0);  c0 = wmma4(a1, bw1[1], c0);
    c1 = wmma4(a0, bw1[2], c1);  c1 = wmma4(a1, bw1[3], c1);
  }

  // silu in C-layout (v_rcp_f32, not IEEE div), spill h(16x32) to LDS slab
  float* const slab = &sScr[wave][0];
#pragma unroll
  for (int rr = 0; rr < 8; ++rr) {
    const int row = rr + (hi ? 8 : 0);
    float v0 = c0[rr]; v0 = v0 * frcp(1.f + __expf(-v0));
    float v1 = c1[rr]; v1 = v1 * frcp(1.f + __expf(-v1));
    slab[row * 32 + m]      = v0;
    slab[row * 32 + m + 16] = v1;
  }
  __syncthreads();

  // ---- MLP layer 2: h(16x32) @ W2(32x16) ----
  v8f cw = {};
#pragma unroll
  for (int ch = 0; ch < 8; ++ch) {
    const v2f ah = *(const v2f*)&slab[m * 32 + ch * 4 + (hi ? 2 : 0)];
    cw = wmma4(ah, bw2[ch], cw);
  }
  __syncthreads();

  // spill w tile (16x16, cols 0..12 valid) and read back per-atom rows
#pragma unroll
  for (int rr = 0; rr < 8; ++rr)
    slab[(rr + (hi ? 8 : 0)) * 16 + m] = cw[rr];
  __syncthreads();

  float wv[13];
  {
    const float4 q0 = *(const float4*)&slab[m * 16 + 0];
    const float4 q1 = *(const float4*)&slab[m * 16 + 4];
    const float4 q2 = *(const float4*)&slab[m * 16 + 8];
    wv[0] = q0.x; wv[1] = q0.y; wv[2]  = q0.z; wv[3]  = q0.w;
    wv[4] = q1.x; wv[5] = q1.y; wv[6]  = q1.z; wv[7]  = q1.w;
    wv[8] = q2.x; wv[9] = q2.y; wv[10] = q2.z; wv[11] = q2.w;
    wv[12] = slab[m * 16 + 12];
  }

  // ---- final contraction: G(16x44) @ T2(44x16), G[e,q]=w[P[q]]*sh[J[q]] ----
  v8f cout = {};
#pragma unroll
  for (int ch = 0; ch < NCHUNK; ++ch) {
    const float g0 = wv[QP[4 * ch + 0]] * shv[QJ[4 * ch + 0]];
    const float g1 = wv[QP[4 * ch + 1]] * shv[QJ[4 * ch + 1]];
    const float g2 = wv[QP[4 * ch + 2]] * shv[QJ[4 * ch + 2]];
    const float g3 = wv[QP[4 * ch + 3]] * shv[QJ[4 * ch + 3]];
    v2f ag; ag.x = hi ? g2 : g0; ag.y = hi ? g3 : g1;
    const v2f bg = ((const v2f*)sT2B)[ch * 32 + lane];
    cout = wmma4(ag, bg, cout);
  }

  // ---- store: C-layout (row = atom, col = output channel < 9) ----
  const int n = m;  // column index for this lane
#pragma unroll
  for (int rr = 0; rr < 8; ++rr) {
    const int eo = tile * 16 + rr + (hi ? 8 : 0);
    if (n < DIMOUT && eo < N) out[eo * DIMOUT + n] = cout[rr];
  }
}

// ---------------------------------------------------------------------------
extern "C" void kernel_launch(void* const* d_in, const int* in_sizes, int n_in,
                              void* d_out, int out_size, void* d_ws,
                              size_t ws_size, hipStream_t stream) {
  const float* f_in = (const float*)d_in[0];
  const float* pos  = (const float*)d_in[1];
  const float* w1   = (const float*)d_in[2];
  const float* w2   = (const float*)d_in[3];
  float* out = (float*)d_out;
  float* ws  = (float*)d_ws;

  const int N = in_sizes[1] / 3;          // pos is (N, 3)
  const int nTiles = (N + 15) / 16;
  const int blocks = (nTiles + 7) / 8;    // 8 waves (tiles) per block

  tp_prep<<<1, 256, 0, stream>>>(f_in, w1, w2, ws);
  if (blocks > 0) tp_conv<<<blocks, 256, 0, stream>>>(pos, ws, out, N);
}